// CrossAttention_Sp_10909216932413
// MI455X (gfx1250) — compile-verified
//
#include <hip/hip_runtime.h>
#include <hip/hip_bf16.h>
#include <math.h>

typedef float v2f __attribute__((ext_vector_type(2)));
typedef float v8f __attribute__((ext_vector_type(8)));

#define BN 8
#define NN 1024
#define DD 256
#define EPSV 1e-8f
#define SSTRIDE 1028  // 1024 + 4-float pad to avoid LDS bank conflicts on column reads

// ---- f32 WMMA wrapper: D = A(16x4) * B(4x16) + C, exact fp32 ----
__device__ __forceinline__ v8f wmma4(v2f a, v2f b, v8f c) {
  return __builtin_amdgcn_wmma_f32_16x16x4_f32(
      /*neg_a=*/false, a, /*neg_b=*/false, b,
      /*c_mod=*/(short)0, c, /*reuse_a=*/false, /*reuse_b=*/false);
}

__device__ __forceinline__ float wave_max(float v) {
#pragma unroll
  for (int off = 16; off > 0; off >>= 1) v = fmaxf(v, __shfl_xor(v, off, 32));
  return v;
}
__device__ __forceinline__ float wave_sum(float v) {
#pragma unroll
  for (int off = 16; off > 0; off >>= 1) v += __shfl_xor(v, off, 32);
  return v;
}

// ---------------- Kernel 1: positional softmax [N,N] ----------------
__global__ void pos_softmax_kernel(const float* __restrict__ coords,
                                   const float* __restrict__ pos_emb,
                                   const float* __restrict__ p_temp,
                                   float* __restrict__ pos_score) {
  __shared__ float row[NN];
  __shared__ float red[8];
  const int n = blockIdx.x;
  const int tid = threadIdx.x;
  const int lane = tid & 31, wave = tid >> 5;
  const float pa = fabsf(p_temp[0]);
  float pe[6];
#pragma unroll
  for (int c = 0; c < 6; ++c) pe[c] = pos_emb[n * 6 + c];

  float lmax = -1e30f;
  for (int m = tid; m < NN; m += 256) {
    const float* cp = coords + ((size_t)n * NN + m) * 6;
    float dot = cp[0] * pe[0] + cp[1] * pe[1] + cp[2] * pe[2] +
                cp[3] * pe[3] + cp[4] * pe[4] + cp[5] * pe[5];
    float lg = -pa * dot;
    row[m] = lg;
    lmax = fmaxf(lmax, lg);
  }
  lmax = wave_max(lmax);
  if (lane == 0) red[wave] = lmax;
  __syncthreads();
  float rmax = red[0];
#pragma unroll
  for (int w = 1; w < 8; ++w) rmax = fmaxf(rmax, red[w]);
  __syncthreads();

  float lsum = 0.f;
  for (int m = tid; m < NN; m += 256) {
    float e = expf(row[m] - rmax);
    row[m] = e;
    lsum += e;
  }
  lsum = wave_sum(lsum);
  if (lane == 0) red[wave] = lsum;
  __syncthreads();
  float rsum = 0.f;
#pragma unroll
  for (int w = 0; w < 8; ++w) rsum += red[w];
  const float inv = 1.0f / rsum;
  for (int m = tid; m < NN; m += 256)
    pos_score[(size_t)n * NN + m] = row[m] * inv;
}

// ---------------- Kernel 2: t = x @ U^T  (M=8192, N=256, K=256) ----------------
// One wave -> one 16x16 tile. Both A and B fragments are contiguous float2 loads.
__global__ void gemm_xut_kernel(const float* __restrict__ x,
                                const float* __restrict__ U,
                                float* __restrict__ t) {
  const int lane = threadIdx.x & 31;
  const int wave = threadIdx.x >> 5;
  const int tile = blockIdx.x * 8 + wave;  // 8192 tiles
  const int i0 = (tile >> 4) * 16;         // 512 row tiles
  const int j0 = (tile & 15) * 16;         // 16 col tiles
  const int hl = lane >> 4, ll = lane & 15;

  const float* arow = x + (size_t)(i0 + ll) * DD + 2 * hl;        // A[m][k..k+1]
  const float* brow = U + (size_t)(j0 + ll) * DD + 2 * hl;        // B[k][n]=U[n][k]
  v8f acc = {0.f, 0.f, 0.f, 0.f, 0.f, 0.f, 0.f, 0.f};
#pragma unroll 8
  for (int kb = 0; kb < DD; kb += 4) {
    v2f a = *(const v2f*)(arow + kb);
    v2f b = *(const v2f*)(brow + kb);
    acc = wmma4(a, b, acc);
  }
  float* orow = t + (size_t)(i0 + 8 * hl) * DD + j0 + ll;
#pragma unroll
  for (int v = 0; v < 8; ++v) orow[(size_t)v * DD] = acc[v];
}

// ---------------- Kernel 3: xs1 = (t*|S1|)@U, xs2 = (t*|S2|)@U ----------------
__global__ void gemm_su_kernel(const float* __restrict__ t,
                               const float* __restrict__ U,
                               const float* __restrict__ S1,
                               const float* __restrict__ S2,
                               float* __restrict__ xs1,
                               float* __restrict__ xs2) {
  const int lane = threadIdx.x & 31;
  const int wave = threadIdx.x >> 5;
  const int tile = blockIdx.x * 8 + wave;
  const int i0 = (tile >> 4) * 16;
  const int j0 = (tile & 15) * 16;
  const int hl = lane >> 4, ll = lane & 15;

  const float* arow = t + (size_t)(i0 + ll) * DD + 2 * hl;
  const float* bcol = U + j0 + ll;  // B[k][n] = U[k*D + j0+n]
  v8f acc1 = {0.f, 0.f, 0.f, 0.f, 0.f, 0.f, 0.f, 0.f};
  v8f acc2 = {0.f, 0.f, 0.f, 0.f, 0.f, 0.f, 0.f, 0.f};
#pragma unroll 4
  for (int kb = 0; kb < DD; kb += 4) {
    const int k = kb + 2 * hl;
    v2f a = *(const v2f*)(arow + kb);
    v2f b;
    b.x = bcol[(size_t)k * DD];
    b.y = bcol[(size_t)(k + 1) * DD];
    v2f a1 = {a.x * fabsf(S1[k]), a.y * fabsf(S1[k + 1])};
    v2f a2 = {a.x * fabsf(S2[k]), a.y * fabsf(S2[k + 1])};
    acc1 = wmma4(a1, b, acc1);
    acc2 = wmma4(a2, b, acc2);
  }
  float* o1 = xs1 + (size_t)(i0 + 8 * hl) * DD + j0 + ll;
  float* o2 = xs2 + (size_t)(i0 + 8 * hl) * DD + j0 + ll;
#pragma unroll
  for (int v = 0; v < 8; ++v) {
    o1[(size_t)v * DD] = acc1[v];
    o2[(size_t)v * DD] = acc2[v];
  }
}

// ---------------- Kernel 4: fused scores -> softmax -> blend -> entropy
//                  -> route -> heat_map -> (attn_sel @ y) ----------------
// One workgroup (8 waves) per (batch, 16-row query tile).
// LDS holds both branches' full score rows: 2*16*1028 floats (~128 KB).
__global__ void fused_attn_kernel(const float* __restrict__ xs1,
                                  const float* __restrict__ xs2,
                                  const float* __restrict__ y,
                                  const float* __restrict__ pos_score,
                                  const float* __restrict__ gating,
                                  const float* __restrict__ h_temp,
                                  float* __restrict__ out,
                                  float* __restrict__ heat) {
  extern __shared__ float smem[];
  float* SL = smem;                       // [2][16][SSTRIDE]
  float* entS = smem + 2 * 16 * SSTRIDE;  // [2][16]
  float* selS = entS + 32;                // [16]

  const int b = blockIdx.x >> 6;  // 64 row-tiles per batch
  const int n0 = (blockIdx.x & 63) * 16;
  const int tid = threadIdx.x;
  const int lane = tid & 31, wave = tid >> 5;
  const int hl = lane >> 4, ll = lane & 15;
  const float scale = 0.0625f;  // D^-0.5
  const float g = 1.0f / (1.0f + expf(-gating[0]));
  const float omg = 1.0f - g;
  const float ht = h_temp[0];
  const float* yb = y + (size_t)b * NN * DD;

  // ---- Pass A: patch logits for both branches into LDS (WMMA f32) ----
  {
    const int branch = wave & 1;
    const float* xs = branch ? xs2 : xs1;
    const float* arow = xs + ((size_t)b * NN + n0 + ll) * DD + 2 * hl;
    for (int mt = (wave >> 1); mt < 64; mt += 4) {  // uniform per wave
      const int m0 = mt * 16;
      const float* brow = yb + (size_t)(m0 + ll) * DD + 2 * hl;  // B[k][m]=y[m][k]
      v8f acc = {0.f, 0.f, 0.f, 0.f, 0.f, 0.f, 0.f, 0.f};
#pragma unroll 8
      for (int kb = 0; kb < DD; kb += 4) {
        v2f a = *(const v2f*)(arow + kb);
        v2f bb = *(const v2f*)(brow + kb);
        acc = wmma4(a, bb, acc);
      }
      float* dst = SL + ((size_t)branch * 16 + 8 * hl) * SSTRIDE + m0 + ll;
#pragma unroll
      for (int v = 0; v < 8; ++v) dst[(size_t)v * SSTRIDE] = acc[v] * scale;
    }
  }
  __syncthreads();

  // ---- Pass B: per-(branch,row) softmax, blend with pos, entropy; in-place ----
  for (int idx = wave; idx < 32; idx += 8) {
    const int br = idx >> 4, row = idx & 15;
    float* Lrow = SL + (size_t)(br * 16 + row) * SSTRIDE;
    const float* prow = pos_score + (size_t)(n0 + row) * NN;
    float lmax = -1e30f;
    for (int m = lane; m < NN; m += 32) lmax = fmaxf(lmax, Lrow[m]);
    lmax = wave_max(lmax);
    float lsum = 0.f;
    for (int m = lane; m < NN; m += 32) lsum += expf(Lrow[m] - lmax);
    lsum = wave_sum(lsum);
    const float inv = 1.0f / lsum;
    float ent = 0.f;
    for (int m = lane; m < NN; m += 32) {
      float p = expf(Lrow[m] - lmax) * inv;
      float a = omg * p + g * prow[m];
      ent += -a * logf(a + EPSV);
      Lrow[m] = a;  // final blended attention value
    }
    ent = wave_sum(ent);
    if (lane == 0) entS[br * 16 + row] = ent;
  }
  __syncthreads();

  // ---- Pass C: routing + heat_map ----
  if (tid < 16) {
    const int row = tid;
    const float h0 = 2.0f - 2.0f / (1.0f + expf(-ht * entS[row]));
    const float h1 = 2.0f - 2.0f / (1.0f + expf(-ht * entS[16 + row]));
    const int sel = (h0 >= h1) ? 0 : 1;
    selS[row] = (float)sel;
    heat[(size_t)b * NN + n0 + row] = sel ? h1 : h0;
  }
  __syncthreads();

  // ---- Pass D: out[n0..n0+15, :] = attn_sel @ y  (K = 1024, WMMA f32) ----
  const int selrow = (int)selS[ll];  // A-fragment row = ll, per-row branch select
  const float* Abase = SL + (size_t)(selrow * 16 + ll) * SSTRIDE + 2 * hl;
  for (int dt = wave; dt < 16; dt += 8) {  // uniform per wave
    const int d0 = dt * 16;
    const float* bbase = yb + d0 + ll;  // B[m][d] = y[m*D + d]
    v8f acc = {0.f, 0.f, 0.f, 0.f, 0.f, 0.f, 0.f, 0.f};
#pragma unroll 4
    for (int kb = 0; kb < NN; kb += 4) {
      const int k = kb + 2 * hl;
      v2f a = *(const v2f*)(Abase + kb);
      v2f bb;
      bb.x = bbase[(size_t)k * DD];
      bb.y = bbase[(size_t)(k + 1) * DD];
      acc = wmma4(a, bb, acc);
    }
    float* orow = out + ((size_t)b * NN + n0 + 8 * hl) * DD + d0 + ll;
#pragma unroll
    for (int v = 0; v < 8; ++v) orow[(size_t)v * DD] = acc[v];
  }
}

extern "C" void kernel_launch(void* const* d_in, const int* in_sizes, int n_in,
                              void* d_out, int out_size, void* d_ws, size_t ws_size,
                              hipStream_t stream) {
  const float* x = (const float*)d_in[0];
  const float* y = (const float*)d_in[1];
  const float* coords = (const float*)d_in[2];
  const float* U = (const float*)d_in[3];
  const float* S1 = (const float*)d_in[4];
  const float* S2 = (const float*)d_in[5];
  const float* gating = (const float*)d_in[6];
  const float* h_temp = (const float*)d_in[7];
  const float* p_temp = (const float*)d_in[8];
  const float* pos_emb = (const float*)d_in[9];

  float* out = (float*)d_out;                       // [B,N,D]
  float* heat = out + (size_t)BN * NN * DD;         // [B,N,1]

  float* ws = (float*)d_ws;
  float* pos_score = ws;                            //  4 MB [N,N]
  float* t = pos_score + (size_t)NN * NN;           //  8 MB [B,N,D]
  float* xs1 = t + (size_t)BN * NN * DD;            //  8 MB
  float* xs2 = xs1 + (size_t)BN * NN * DD;          //  8 MB  (28 MB total)

  pos_softmax_kernel<<<NN, 256, 0, stream>>>(coords, pos_emb, p_temp, pos_score);
  gemm_xut_kernel<<<1024, 256, 0, stream>>>(x, U, t);
  gemm_su_kernel<<<1024, 256, 0, stream>>>(t, U, S1, S2, xs1, xs2);

  const size_t lds_bytes = (size_t)(2 * 16 * SSTRIDE + 32 + 16) * sizeof(float);
  fused_attn_kernel<<<BN * 64, 256, lds_bytes, stream>>>(
      xs1, xs2, y, pos_score, gating, h_temp, out, heat);
}